// AutoCorrelation_78048145703109
// MI455X (gfx1250) — compile-verified
//
#include <hip/hip_runtime.h>
#include <hip/hip_bf16.h>

// Problem constants from the reference: B=16, H=16, S=2048, D=64, fp32.
constexpr int kS  = 2048;   // sequence length
constexpr int kD  = 64;     // head dim
constexpr int kBH = 256;    // B*H

typedef float v2f __attribute__((ext_vector_type(2)));
typedef float v8f __attribute__((ext_vector_type(8)));
typedef unsigned int v4u __attribute__((ext_vector_type(4)));
typedef int v4i __attribute__((ext_vector_type(4)));
typedef int v8i __attribute__((ext_vector_type(8)));

// ---------------------------------------------------------------------------
// CDNA5 data-movement helpers
// ---------------------------------------------------------------------------

// Generic LDS pointer = {shared_aperture_hi32, lds_byte_offset}; the low 32
// bits ARE the LDS byte address the hardware wants (ISA 10.2 aperture rules).
__device__ __forceinline__ unsigned lds_off(const void* p) {
    return (unsigned)(unsigned long long)p;
}

// Per-lane async copy global -> LDS (tracked on ASYNCcnt, no VGPR round trip).
__device__ __forceinline__ void async_load_b32_to_lds(unsigned lds_byte, const void* g) {
    asm volatile("global_load_async_to_lds_b32 %0, %1, off"
                 :: "v"(lds_byte), "v"((unsigned long long)g)
                 : "memory");
}

__device__ __forceinline__ void wait_asynccnt0() {
    asm volatile("s_wait_asynccnt 0x0" ::: "memory");
}

// Tensor Data Mover: DMA one 2D fp32 tile (tile_w x tile_h) from a row-major
// tensor (row stride row_stride elems) into LDS, inserting pad_amount dwords
// of padding every pad_interval dwords (codes per D# group1: interval code c
// => 2^(c+1) dwords; amount code a => a+1 dwords).
// D# layout per CDNA5 ISA 8.3/8.4: group0 {count=1, lds_addr, global_addr,
// type=2}, group1 {data_size=4B, pad ctl, tensor dims/strides, tile dims},
// groups 2/3 (and the trailing group on this toolchain) zero: tile_dim2=0
// makes the tile 2D so the higher-dimension descriptors are unused.
__device__ __forceinline__ void tdm_load_2d(unsigned lds_addr, const void* gptr,
                                            unsigned tensor_w, unsigned tensor_h,
                                            unsigned row_stride,
                                            unsigned tile_w, unsigned tile_h,
                                            unsigned pad_interval_code,
                                            unsigned pad_amount_code)
{
    const unsigned long long ga = (unsigned long long)gptr;
    v4u g0;
    g0[0] = 1u;                                   // count=1, user mode, no gather
    g0[1] = lds_addr;                             // LDS byte address
    g0[2] = (unsigned)ga;                         // global_addr[31:0]
    g0[3] = (unsigned)((ga >> 32) & 0x01FFFFFFull) | (2u << 30);  // [56:32] | type=2

    v8i g1;
    g1[0] = (int)((2u << 16)                      // data_size = 4 bytes
                | (1u << 20)                      // pad_enable
                | (pad_interval_code << 22)
                | (pad_amount_code << 25));       // workgroup_mask = 0
    g1[1] = (int)((tensor_w & 0xFFFFu) << 16);    // tensor_dim0[15:0]
    g1[2] = (int)(((tensor_w >> 16) & 0xFFFFu) | ((tensor_h & 0xFFFFu) << 16));
    g1[3] = (int)(((tensor_h >> 16) & 0xFFFFu) | ((tile_w & 0xFFFFu) << 16));
    g1[4] = (int)(tile_h & 0xFFFFu);              // tile_dim1; tile_dim2=0 (2D)
    g1[5] = (int)row_stride;                      // tensor_dim0_stride[31:0]
    g1[6] = 0;                                    // stride0 hi / stride1 lo
    g1[7] = 0;

    const v4i z4 = {0, 0, 0, 0};
    const v8i z8 = {0, 0, 0, 0, 0, 0, 0, 0};
    // clang-23 / therock-10.0 lane: 6-arg form (extra int32x8 group + cpol).
    __builtin_amdgcn_tensor_load_to_lds(g0, g1, z4, z4, z8, 0);
}

// ---------------------------------------------------------------------------
// Kernel 1: one workgroup per (b,h,d) column.
//   - circular cross-correlation corr[n] = sum_t q[t]*k[(t-n) mod S]
//   - full descending bitonic sort of the 2048 lags in LDS
//   - softmax over the sorted axis
//   - write normalized weights to ws in (bh, d, t) layout (GEMM-friendly)
// ---------------------------------------------------------------------------
__global__ __launch_bounds__(256)
void corr_sort_softmax_kernel(const float* __restrict__ q,
                              const float* __restrict__ kmat,
                              float* __restrict__ w)   // (kBH, kD, kS)
{
    __shared__ float qs[kS];
    __shared__ float ks[2 * kS];   // doubled to avoid the mod in the inner loop
    __shared__ float cs[kS];
    __shared__ float red[256];

    const int col = blockIdx.x;    // 0 .. kBH*kD-1
    const int bh  = col >> 6;
    const int d   = col & 63;
    const int tid = threadIdx.x;

    const float* qp = q    + (size_t)bh * kS * kD + d;
    const float* kp = kmat + (size_t)bh * kS * kD + d;

    // Stage the two columns straight into LDS with async copies (stride-D
    // gather; reuse across the 64 columns of one (b,h) lives in the 192MB L2).
    for (int t = tid; t < kS; t += 256) {
        const float* qg = qp + (size_t)t * kD;
        const float* kg = kp + (size_t)t * kD;
        async_load_b32_to_lds(lds_off(&qs[t]),      qg);
        async_load_b32_to_lds(lds_off(&ks[t]),      kg);
        async_load_b32_to_lds(lds_off(&ks[t + kS]), kg);
    }
    wait_asynccnt0();
    __syncthreads();

    // Each thread accumulates 8 lags: n = tid + 256*i.
    float acc[8];
#pragma unroll
    for (int i = 0; i < 8; ++i) acc[i] = 0.0f;

    for (int t0 = 0; t0 < kS; t0 += 8) {
        float qr[8];
#pragma unroll
        for (int j = 0; j < 8; ++j) qr[j] = qs[t0 + j];
#pragma unroll
        for (int i = 0; i < 8; ++i) {
            const int n = tid + (i << 8);
            const float* kb = &ks[t0 - n + kS];  // index in [1, 4095]
#pragma unroll
            for (int j = 0; j < 8; ++j) acc[i] = fmaf(qr[j], kb[j], acc[i]);
        }
    }

#pragma unroll
    for (int i = 0; i < 8; ++i) cs[tid + (i << 8)] = acc[i];
    __syncthreads();

    // Bitonic sort, descending, 2048 elements, 256 threads (4 CE/thread/pass).
    for (int kk = 2; kk <= kS; kk <<= 1) {
        for (int j = kk >> 1; j > 0; j >>= 1) {
            for (int p = tid; p < kS / 2; p += 256) {
                const int i  = ((p & ~(j - 1)) << 1) | (p & (j - 1));
                const int ix = i | j;
                const float a = cs[i];
                const float b = cs[ix];
                const bool up = (i & kk) == 0;
                const bool sw = up ? (a < b) : (a > b);
                if (sw) { cs[i] = b; cs[ix] = a; }
            }
            __syncthreads();
        }
    }

    // Softmax over the sorted axis; max is element 0 after descending sort.
    const float mx = cs[0];
    __syncthreads();

    float ls = 0.0f;
    for (int t = tid; t < kS; t += 256) {
        const float e = __expf(cs[t] - mx);
        cs[t] = e;
        ls += e;
    }
    red[tid] = ls;
    __syncthreads();
    for (int sh = 128; sh > 0; sh >>= 1) {
        if (tid < sh) red[tid] += red[tid + sh];
        __syncthreads();
    }
    const float inv = 1.0f / red[0];

    float* wp = w + (size_t)col * kS;
    for (int t = tid; t < kS; t += 256) wp[t] = cs[t] * inv;
}

// ---------------------------------------------------------------------------
// Kernel 2: out[bh, t, y] = sum_x W[bh, t, x] * V[bh, x, y]   (x,y < 64)
// W stored transposed as (bh, x, t) in ws. Per (b,h): 2048x64x64 GEMM.
// WMMA f32 16x16x4: each wave owns one 16(t)x16(y) C tile, K-loop of 16.
// Block = 8 waves, walks 4 consecutive 32-row t-blocks (strip of 128 rows).
// V tile (64x64) + double-buffered W slabs (64x32) are DMA'd by the Tensor
// Data Mover; the next slab's DMA overlaps the current WMMA loop.
// ---------------------------------------------------------------------------
__global__ __launch_bounds__(256)
void weighted_gemm_kernel(const float* __restrict__ w,   // (kBH, kD, kS)
                          const float* __restrict__ v,   // (kBH, kS, kD)
                          float* __restrict__ out)       // (kBH, kS, kD)
{
    __shared__ float Vs[kD][kD + 4];     // rows padded by TDM (64+4 dwords)
    __shared__ float Ws[2][kD][32 + 4];  // rows padded by TDM (32+4 dwords)

    const int blk   = blockIdx.x;
    const int bh    = blk >> 4;          // 16 strips of 128 rows per (b,h)
    const int strip = blk & 15;
    const int tbase = strip * 128;
    const int tid   = threadIdx.x;

    const float* vbase = v + (size_t)bh * kS * kD;
    const float* wbase = w + (size_t)bh * kD * kS;

    if (tid == 0) {
        // V tile: 64x64, contiguous rows (stride 64); pad 64dw->+4dw (codes 5,3)
        tdm_load_2d(lds_off(&Vs[0][0]), vbase, kD, kD, kD, kD, kD, 5u, 3u);
        // W slab 0: 32 wide (t), 64 rows (x), row stride 2048; pad 32dw->+4dw
        tdm_load_2d(lds_off(&Ws[0][0][0]), wbase + tbase, kS, kD, kS, 32u, kD, 4u, 3u);
    }
    __builtin_amdgcn_s_wait_tensorcnt(0);
    __syncthreads();

    const int wave  = tid >> 5;      // 8 waves (wave32)
    const int lane  = tid & 31;
    const int ttile = wave >> 2;     // 0..1 : 16-row t tile within the 32-block
    const int ytile = wave & 3;      // 0..3 : 16-col y tile
    const int m     = lane & 15;     // A row / B,C column index
    const int half  = lane >> 4;     // 0: K=0,1 & M=r ; 1: K=2,3 & M=r+8
    const int kb    = half * 2;

    for (int it = 0; it < 4; ++it) {
        if (tid == 0 && it < 3) {    // prefetch next W slab while we compute
            tdm_load_2d(lds_off(&Ws[(it + 1) & 1][0][0]),
                        wbase + tbase + (it + 1) * 32, kS, kD, kS, 32u, kD, 4u, 3u);
        }

        const float (*Wb)[32 + 4] = Ws[it & 1];

        v8f acc = {};
#pragma unroll
        for (int kstep = 0; kstep < 16; ++kstep) {
            const int x0 = kstep * 4;
            v2f a, b;
            // A 16x4 f32 layout: lanes 0-15 hold K=0,1; lanes 16-31 hold K=2,3
            a.x = Wb[x0 + kb + 0][ttile * 16 + m];
            a.y = Wb[x0 + kb + 1][ttile * 16 + m];
            // B 4x16 f32 layout (mirrors A/C halves)
            b.x = Vs[x0 + kb + 0][ytile * 16 + m];
            b.y = Vs[x0 + kb + 1][ytile * 16 + m];
            acc = __builtin_amdgcn_wmma_f32_16x16x4_f32(false, a, false, b,
                                                        (short)0, acc, false, false);
        }

        // C/D layout: VGPR r -> M=r (lanes 0-15) / M=r+8 (lanes 16-31).
        const int row0 = tbase + it * 32 + ttile * 16 + half * 8;
        float* op = out + ((size_t)bh * kS + row0) * kD + ytile * 16 + m;
#pragma unroll
        for (int r = 0; r < 8; ++r) {
            op[(size_t)r * kD] = acc[r];
        }

        // Drain the prefetch DMA (wave 0) before anyone touches the other
        // buffer; barrier also fences buffer reuse two iterations out.
        __builtin_amdgcn_s_wait_tensorcnt(0);
        __syncthreads();
    }
}

// ---------------------------------------------------------------------------
extern "C" void kernel_launch(void* const* d_in, const int* in_sizes, int n_in,
                              void* d_out, int out_size, void* d_ws, size_t ws_size,
                              hipStream_t stream) {
    const float* q   = (const float*)d_in[0];   // queries (B,H,S,D)
    const float* k   = (const float*)d_in[1];   // keys    (B,H,S,D)
    const float* v   = (const float*)d_in[2];   // values  (B,H,S,D)
    float*       out = (float*)d_out;           // (B,H,S,D)
    float*       wgt = (float*)d_ws;            // needs kBH*kD*kS*4 = 128 MB

    (void)in_sizes; (void)n_in; (void)out_size; (void)ws_size;

    corr_sort_softmax_kernel<<<kBH * kD, 256, 0, stream>>>(q, k, wgt);
    weighted_gemm_kernel<<<kBH * (kS / 128), 256, 0, stream>>>(wgt, v, out);
}